// Model_59682865545775
// MI455X (gfx1250) — compile-verified
//
#include <hip/hip_runtime.h>

#define B_TOT 8192
#define T_LEN 365
#define IN_F  5
#define HID   10

typedef float v2f __attribute__((ext_vector_type(2)));
typedef float v8f __attribute__((ext_vector_type(8)));

__device__ __forceinline__ float fsig(float x) {
  // 1/(1+exp(-x)) via v_exp2 + v_rcp
  float e = __builtin_amdgcn_exp2f(-1.4426950408889634f * x);
  return __builtin_amdgcn_rcpf(1.0f + e);
}
__device__ __forceinline__ float ftanh(float x) {
  // tanh(x) = 2/(1+exp(-2x)) - 1
  float e = __builtin_amdgcn_exp2f(-2.8853900817779268f * x);
  return __builtin_amdgcn_rcpf(1.0f + e) * 2.0f - 1.0f;
}

// One wave (32 lanes) carries 16 batch elements through all 365 timesteps.
// A (weights) = [Wih;Whh]^T : 48x16 (3 gate-tiles x 4 K-chunks), loaded once.
// Gate-row permutation: tile tI, row m -> unit u=4*tI+(m>>2), gate ty=m&3 (f,i,o,g),
// so each lane owns all 4 gates of its units after WMMA (lane-local activations).
__global__ __launch_bounds__(128)
void lstm_wmma_kernel(const float* __restrict__ x,
                      const float* __restrict__ wih,
                      const float* __restrict__ whh,
                      const float* __restrict__ bias,
                      const float* __restrict__ fcw,
                      const float* __restrict__ fcb,
                      float* __restrict__ out) {
  const int lane = threadIdx.x & 31;
  const int wid  = blockIdx.x * (blockDim.x >> 5) + (threadIdx.x >> 5);
  const int half = lane >> 4;      // 0: rows M0-7 of D, 1: rows M8-15
  const int n    = lane & 15;      // batch column within tile
  const int b    = wid * 16 + n;

  float* outv = out;                                   // [B]
  float* hseq = out + B_TOT;                           // [B,T,H]
  float* cseq = hseq + (size_t)B_TOT * T_LEN * HID;    // [B,T,H]

  // ---- A fragments (constant over time). A[m][k], m = lane&15.
  v2f A[3][4];
  {
    const int m = n;
    #pragma unroll
    for (int tI = 0; tI < 3; ++tI) {
      const int u   = 4 * tI + (m >> 2);
      const int ty  = m & 3;
      const int col = ty * 10 + u;       // original gate column (f,i,o,g blocks of 10)
      const bool pad = (u >= HID);
      #pragma unroll
      for (int j = 0; j < 4; ++j) {
        #pragma unroll
        for (int e = 0; e < 2; ++e) {
          const int k = 4 * j + e + 2 * half;   // lanes16-31 carry K+2
          float v = 0.0f;
          if (!pad && k < 15)
            v = (k < IN_F) ? wih[k * 40 + col] : whh[(k - IN_F) * 40 + col];
          A[tI][j][e] = v;
        }
      }
    }
  }

  // ---- bias pre-staged in C layout (row m = r + 8*half)
  v8f biasC[3];
  #pragma unroll
  for (int tI = 0; tI < 3; ++tI) {
    #pragma unroll
    for (int r = 0; r < 8; ++r) {
      const int m = r + 8 * half;
      const int u = 4 * tI + (m >> 2);
      biasC[tI][r] = (u < HID) ? bias[(m & 3) * 10 + u] : 0.0f;
    }
  }

  // Recurrent state. Lower lanes own units {0,1,4,5,8,9}, upper own {2,3,6,7}.
  float hs[6] = {0,0,0,0,0,0};
  float cs[6] = {0,0,0,0,0,0};

  const float* xrow = x    + (size_t)b * T_LEN * IN_F;
  float*       hrow = hseq + (size_t)b * T_LEN * HID;
  float*       crow = cseq + (size_t)b * T_LEN * HID;

  for (int t = 0; t < T_LEN; ++t) {
    const float* xt = xrow + t * IN_F;

    // ---- Build B (16x16 input column-matrix [x_t ; h_t ; 0]), 4 K-chunks.
    v2f Bf[4];
    Bf[0][0] = xt[half ? 2 : 0];          // rows k=0 / k=2
    Bf[0][1] = xt[half ? 3 : 1];          // rows k=1 / k=3
    {
      float x4 = xt[4];
      float g1 = __shfl(hs[1], n, 32);            // h1 (owner: lower, slot1)
      Bf[1][0] = half ? g1 : x4;                  // k=4 (x) / k=6 (h1)
      Bf[1][1] = hs[0];                           // k=5 (h0) / k=7 (h2): same-half
    }
    {
      float g3 = __shfl(hs[1], n + 16, 32);       // h3 (owner: upper, slot1)
      float g5 = __shfl(hs[3], n, 32);            // h5 (owner: lower, slot3)
      Bf[2][0] = half ? g5 : g3;                  // k=8 (h3) / k=10 (h5)
      Bf[2][1] = hs[2];                           // k=9 (h4) / k=11 (h6): same-half
    }
    {
      float g7 = __shfl(hs[3], n + 16, 32);       // h7 (owner: upper, slot3)
      float g9 = __shfl(hs[5], n, 32);            // h9 (owner: lower, slot5)
      Bf[3][0] = half ? g9 : g7;                  // k=12 (h7) / k=14 (h9)
      Bf[3][1] = half ? 0.0f : hs[4];             // k=13 (h8) / k=15 (pad)
    }

    // ---- gates = A(48x16-ish) x B(16x16) + bias : 12 x v_wmma_f32_16x16x4_f32
    v8f acc0 = biasC[0], acc1 = biasC[1], acc2 = biasC[2];
    #pragma unroll
    for (int j = 0; j < 4; ++j) {
      acc0 = __builtin_amdgcn_wmma_f32_16x16x4_f32(false, A[0][j], false, Bf[j],
                                                   (short)0, acc0, false, false);
      acc1 = __builtin_amdgcn_wmma_f32_16x16x4_f32(false, A[1][j], false, Bf[j],
                                                   (short)0, acc1, false, false);
      acc2 = __builtin_amdgcn_wmma_f32_16x16x4_f32(false, A[2][j], false, Bf[j],
                                                   (short)0, acc2, false, false);
    }

    // ---- lane-local LSTM cell update (slot s: tile s/2, regs 4*(s&1)..+3)
    auto update = [&](float f, float i, float o, float g, int s) {
      float c1 = fsig(f) * cs[s] + fsig(i) * ftanh(g);
      float h1 = fsig(o) * ftanh(c1);
      cs[s] = c1; hs[s] = h1;
    };
    update(acc0[0], acc0[1], acc0[2], acc0[3], 0);
    update(acc0[4], acc0[5], acc0[6], acc0[7], 1);
    update(acc1[0], acc1[1], acc1[2], acc1[3], 2);
    update(acc1[4], acc1[5], acc1[6], acc1[7], 3);
    update(acc2[0], acc2[1], acc2[2], acc2[3], 4);
    update(acc2[4], acc2[5], acc2[6], acc2[7], 5);

    // ---- stream h,c to global (unit pairs are contiguous -> float2 stores)
    const int u0 = 2 * half;
    float2 v;
    v.x = hs[0]; v.y = hs[1]; *(float2*)(hrow + t * HID + u0)     = v;
    v.x = cs[0]; v.y = cs[1]; *(float2*)(crow + t * HID + u0)     = v;
    v.x = hs[2]; v.y = hs[3]; *(float2*)(hrow + t * HID + u0 + 4) = v;
    v.x = cs[2]; v.y = cs[3]; *(float2*)(crow + t * HID + u0 + 4) = v;
    if (!half) {
      v.x = hs[4]; v.y = hs[5]; *(float2*)(hrow + t * HID + 8) = v;
      v.x = cs[4]; v.y = cs[5]; *(float2*)(crow + t * HID + 8) = v;
    }
  }

  // ---- final FC on h_{T-1}: partial dot per lane, combine the lane pair
  float partial = 0.0f;
  #pragma unroll
  for (int s = 0; s < 6; ++s) {
    int u = 4 * (s >> 1) + 2 * half + (s & 1);
    if (u < HID) partial += hs[s] * fcw[u];
  }
  partial += __shfl_xor(partial, 16, 32);
  if (!half) outv[b] = partial + fcb[0];
}

extern "C" void kernel_launch(void* const* d_in, const int* in_sizes, int n_in,
                              void* d_out, int out_size, void* d_ws, size_t ws_size,
                              hipStream_t stream) {
  (void)in_sizes; (void)n_in; (void)out_size; (void)d_ws; (void)ws_size;
  const float* x    = (const float*)d_in[0];
  const float* wih  = (const float*)d_in[1];
  const float* whh  = (const float*)d_in[2];
  const float* bias = (const float*)d_in[3];
  const float* fcw  = (const float*)d_in[4];
  const float* fcb  = (const float*)d_in[5];
  float* out = (float*)d_out;

  dim3 block(128);                       // 4 waves of 32
  dim3 grid(B_TOT / 16 / 4);             // 512 waves total, 16 batch rows each
  lstm_wmma_kernel<<<grid, block, 0, stream>>>(x, wih, whh, bias, fcw, fcb, out);
}